// CPAB_AR_53249004536361
// MI455X (gfx1250) — compile-verified
//
#include <hip/hip_runtime.h>
#include <hip/hip_bf16.h>

// CPAB autoregressive flow for MI455X (gfx1250, wave32, WMMA).
//
// Roofline: ~8.5 GFLOP fp32 GEMM + ~2 G scalar-VALU ODE ops vs ~48 MB HBM
// traffic (~2us @ 23.3 TB/s) -> compute bound. GEMMs go to
// v_wmma_f32_16x16x4_f32 (exact fp32, matches reference precision); the
// 11-step ODE runs as coalesced scalar VALU work over all 16*B elements.

typedef __attribute__((ext_vector_type(2))) float v2f;
typedef __attribute__((ext_vector_type(8))) float v8f;

#define DIMX  16
#define SX    15
#define HIDX  10
#define NHLX  3
#define DTH   9
#define WAVES 8
#define BLKT  (WAVES * 32)

__device__ __forceinline__ v8f wmma_f32_k4(v2f a, v2f b, v8f c) {
  // D = A(16x4 f32) * B(4x16 f32) + C(16x16 f32); 8-arg form:
  // (neg_a, A, neg_b, B, c_mod, C, reuse_a, reuse_b)
  return __builtin_amdgcn_wmma_f32_16x16x4_f32(false, a, false, b,
                                               (short)0, c, false, false);
}

// Intra-wave LDS ordering fence (CDNA5 split counters): DS ops from one wave
// are in-order; waiting DScnt==0 makes cross-lane LDS writes visible.
__device__ __forceinline__ void lds_wave_fence() {
  asm volatile("s_wait_dscnt 0" ::: "memory");
}

__global__ __launch_bounds__(BLKT, 1)
void cpab_ar_kernel(const float* __restrict__ x,
                    const float* __restrict__ init_param,
                    const float* __restrict__ W0,  const float* __restrict__ b0,
                    const float* __restrict__ Wh,  const float* __restrict__ bh,
                    const float* __restrict__ Wout,const float* __restrict__ bout,
                    float* __restrict__ z_out, float* __restrict__ ld_out,
                    int nrows)
{
  // WMMA-fragment-friendly, zero-padded weight staging.
  // B-fragment for k-step ks, lane half h, col n  ==  v2f at [2*ks+h][n].
  __shared__ v2f   s_w0[SX * 8 * 16];            // K padded 15->16
  __shared__ v2f   s_wh[SX * NHLX * 6 * 16];     // K padded 10->12
  __shared__ v2f   s_wo[SX * 6 * 16];            // K padded 10->12, N 9->16
  __shared__ float s_b0[SX * 16];
  __shared__ float s_bh[SX * NHLX * 16];
  __shared__ float s_bo[SX * 16];
  __shared__ float s_theta[WAVES][16 * 16 * DTH]; // [row][dim][9], stride 9 (odd)
  __shared__ float s_scr[WAVES][16 * 17];         // D->A transpose, padded row

  const int tid  = threadIdx.x;
  const int wave = tid >> 5;
  const int lane = tid & 31;
  const int m    = lane & 15;   // row (A) or col (B/C) index
  const int half = lane >> 4;   // K-half selector

  // ---- stage weights into LDS (zero the padded tiles first) ----
  {
    float* fw0 = (float*)s_w0;
    float* fwh = (float*)s_wh;
    float* fwo = (float*)s_wo;
    v2f z2; z2.x = 0.f; z2.y = 0.f;
    for (int i = tid; i < SX*8*16;        i += BLKT) s_w0[i] = z2;
    for (int i = tid; i < SX*NHLX*6*16;   i += BLKT) s_wh[i] = z2;
    for (int i = tid; i < SX*6*16;        i += BLKT) s_wo[i] = z2;
    for (int i = tid; i < SX*16;          i += BLKT) { s_b0[i] = 0.f; s_bo[i] = 0.f; }
    for (int i = tid; i < SX*NHLX*16;     i += BLKT) s_bh[i] = 0.f;
    __syncthreads();

    for (int idx = tid; idx < SX*SX*HIDX; idx += BLKT) {        // W0 [15][15][10]
      int n = idx % 10, k = (idx / 10) % 15, i = idx / 150;
      fw0[((i*8 + (k >> 1))*16 + n)*2 + (k & 1)] = W0[idx];
    }
    for (int idx = tid; idx < SX*NHLX*HIDX*HIDX; idx += BLKT) { // Wh [15][3][10][10]
      int n = idx % 10, k = (idx / 10) % 10, j = (idx / 100) % 3, i = idx / 300;
      fwh[(((i*3 + j)*6 + (k >> 1))*16 + n)*2 + (k & 1)] = Wh[idx];
    }
    for (int idx = tid; idx < SX*HIDX*DTH; idx += BLKT) {       // Wout [15][10][9]
      int n = idx % 9, k = (idx / 9) % 10, i = idx / 90;
      fwo[((i*6 + (k >> 1))*16 + n)*2 + (k & 1)] = Wout[idx];
    }
    for (int idx = tid; idx < SX*HIDX; idx += BLKT) {           // b0 [15][10]
      s_b0[(idx / 10)*16 + (idx % 10)] = b0[idx];
    }
    for (int idx = tid; idx < SX*NHLX*HIDX; idx += BLKT) {      // bh [15][3][10]
      int n = idx % 10, j = (idx / 10) % 3, i = idx / 30;
      s_bh[(i*3 + j)*16 + n] = bh[idx];
    }
    for (int idx = tid; idx < SX*DTH; idx += BLKT) {            // bout [15][9]
      s_bo[(idx / 9)*16 + (idx % 9)] = bout[idx];
    }
    __syncthreads();
  }

  const int rowBase = (blockIdx.x * WAVES + wave) * 16;

  // theta for dim 0 = init_param broadcast
  for (int e = lane; e < 16 * DTH; e += 32) {
    int r = e / DTH, c = e % DTH;
    s_theta[wave][(r*16 + 0)*DTH + c] = init_param[c];
  }

  // ---- load & clip this wave's 16x16 x-tile as A fragments ----
  // A 16x4 f32 layout: lanes 0-15 -> K0/K1 in v0/v1, lanes 16-31 -> K2/K3.
  v2f xr[4];
  {
    int rowm = rowBase + m; if (rowm > nrows - 1) rowm = nrows - 1;
    const float* xrow = x + (size_t)rowm * DIMX;
#pragma unroll
    for (int ks = 0; ks < 4; ++ks) {
      int c0 = 4*ks + 2*half;
      v2f t;
      t.x = fminf(fmaxf(xrow[c0],     1e-7f), 1.f - 1e-7f);
      t.y = fminf(fmaxf(xrow[c0 + 1], 1e-7f), 1.f - 1e-7f);
      xr[ks] = t;
    }
  }

  // ---- 15 autoregressive MLPs, 16 batch rows each, on fp32 WMMA ----
#pragma unroll 1
  for (int i = 0; i < SX; ++i) {
    // layer 0: (x .* mask_i)[16x16] @ W0_i[16x16] + b0_i   (no activation)
    v8f c;
    {
      float bias = s_b0[i*16 + m];
#pragma unroll
      for (int g = 0; g < 8; ++g) c[g] = bias;
      const v2f* w0i = s_w0 + i*8*16;
#pragma unroll
      for (int ks = 0; ks < 4; ++ks) {
        int c0 = 4*ks + 2*half;
        v2f a;
        a.x = (c0     <= i) ? xr[ks].x : 0.f;   // autoregressive mask j<=i
        a.y = (c0 + 1 <= i) ? xr[ks].y : 0.f;
        v2f bf = w0i[(2*ks + half)*16 + m];
        c = wmma_f32_k4(a, bf, c);
      }
    }

    // hidden layers: relu(h @ Wh + bh), K padded to 12
#pragma unroll
    for (int j = 0; j < NHLX; ++j) {
      // D -> A transpose via per-wave LDS scratch
#pragma unroll
      for (int g = 0; g < 8; ++g) s_scr[wave][(g + 8*half)*17 + m] = c[g];
      lds_wave_fence();
      float bias = s_bh[(i*3 + j)*16 + m];
      v8f cn;
#pragma unroll
      for (int g = 0; g < 8; ++g) cn[g] = bias;
      const v2f* whij = s_wh + (i*3 + j)*6*16;
#pragma unroll
      for (int ks = 0; ks < 3; ++ks) {
        int c0 = 4*ks + 2*half;
        v2f a;
        a.x = s_scr[wave][m*17 + c0];
        a.y = s_scr[wave][m*17 + c0 + 1];
        v2f bf = whij[(2*ks + half)*16 + m];
        cn = wmma_f32_k4(a, bf, cn);
      }
#pragma unroll
      for (int g = 0; g < 8; ++g) cn[g] = fmaxf(cn[g], 0.f);  // ReLU
      c = cn;
    }

    // output layer: theta_i[16x9] = h @ Wout_i + bout_i
    {
#pragma unroll
      for (int g = 0; g < 8; ++g) s_scr[wave][(g + 8*half)*17 + m] = c[g];
      lds_wave_fence();
      float bias = s_bo[i*16 + m];
      v8f co;
#pragma unroll
      for (int g = 0; g < 8; ++g) co[g] = bias;
      const v2f* woi = s_wo + i*6*16;
#pragma unroll
      for (int ks = 0; ks < 3; ++ks) {
        int c0 = 4*ks + 2*half;
        v2f a;
        a.x = s_scr[wave][m*17 + c0];
        a.y = s_scr[wave][m*17 + c0 + 1];
        v2f bf = woi[(2*ks + half)*16 + m];
        co = wmma_f32_k4(a, bf, co);
      }
      if (m < DTH) {
#pragma unroll
        for (int g = 0; g < 8; ++g) {
          int r = g + 8*half;
          s_theta[wave][(r*16 + (i + 1))*DTH + m] = co[g];
        }
      }
    }
  }

  lds_wave_fence();  // thetas visible to all lanes of this wave

  // ---- CPAB: exact piecewise-affine ODE, 11 steps, per (row,dim) ----
  // lane mapping e = lane + 32q, dim = e&15, row = e>>4 -> coalesced I/O.
#pragma unroll 1
  for (int q = 0; q < 8; ++q) {
    int e = lane + 32*q;
    int d = e & 15;
    int r = e >> 4;
    int rr = rowBase + r;
    int rrc = rr > nrows - 1 ? nrows - 1 : rr;
    float xc = x[(size_t)rrc * DIMX + d];
    xc = fminf(fmaxf(xc, 1e-7f), 1.f - 1e-7f);
    const float* th = &s_theta[wave][(r*16 + d)*DTH];
    float t = 1.f, ld = 0.f;
#pragma unroll 1
    for (int it = 0; it < 11; ++it) {
      int cell = (int)floorf(xc * 10.f);
      cell = cell < 0 ? 0 : (cell > 9 ? 9 : cell);
      float vlo = th[cell >= 1 ? cell - 1 : 0];
      float vhi = th[cell <= 8 ? cell : 8];
      float vc  = (cell >= 1) ? vlo : 0.f;     // v[cell]
      float vc1 = (cell <= 8) ? vhi : 0.f;     // v[cell+1]
      float ac = (vc1 - vc) * 10.f;
      float bc = vc - ac * (0.1f * (float)cell);
      float vel = fmaf(ac, xc, bc);
      float xb  = (vel >= 0.f) ? 0.1f * (float)(cell + 1) : 0.1f * (float)cell;
      float vb  = fmaf(ac, xb, bc);
      bool  small   = fabsf(ac) < 1e-8f;
      float velsafe = (vel == 0.f) ? 1.f : vel;
      float acsafe  = small ? 1.f : ac;
      float dt_lin  = (xb - xc) / velsafe;
      float ratio   = fmaxf(vb / velsafe, 1e-38f);
      float dt_exp  = __logf(ratio) / acsafe;
      float dt_hit  = small ? dt_lin : dt_exp;
      dt_hit = ((vel == 0.f) || (dt_hit < 0.f)) ? __builtin_inff() : dt_hit;
      float dt    = fminf(t, dt_hit);
      float x_lin = fmaf(vel, dt, xc);
      float boa   = bc / acsafe;
      float x_exp = fmaf(xc + boa, __expf(ac * dt), -boa);
      float x_new = small ? x_lin : x_exp;
      bool  hit   = dt_hit <= t;
      x_new = hit ? (xb + (vel > 0.f ? 1e-8f : -1e-8f)) : x_new;
      xc = fminf(fmaxf(x_new, 0.f), 1.f);
      t  = fmaxf(t - dt, 0.f);
      ld = fmaf(ac, dt, ld);
    }
    if (rr < nrows) {
      z_out[(size_t)rr * DIMX + d]  = xc;
      ld_out[(size_t)rr * DIMX + d] = ld;
    }
  }
}

extern "C" void kernel_launch(void* const* d_in, const int* in_sizes, int n_in,
                              void* d_out, int out_size, void* d_ws, size_t ws_size,
                              hipStream_t stream) {
  (void)n_in; (void)out_size; (void)d_ws; (void)ws_size;
  const float* x    = (const float*)d_in[0];
  const float* ip   = (const float*)d_in[1];
  const float* W0   = (const float*)d_in[2];
  const float* b0   = (const float*)d_in[3];
  const float* Wh   = (const float*)d_in[4];
  const float* bh   = (const float*)d_in[5];
  const float* Wout = (const float*)d_in[6];
  const float* bout = (const float*)d_in[7];
  int nrows = in_sizes[0] / DIMX;            // B
  float* z  = (float*)d_out;                 // z [B][16]
  float* ld = z + (size_t)nrows * DIMX;      // log-grad [B][16]
  int grid = (nrows + 16 * WAVES - 1) / (16 * WAVES);
  cpab_ar_kernel<<<grid, BLKT, 0, stream>>>(x, ip, W0, b0, Wh, bh, Wout, bout,
                                            z, ld, nrows);
}